// TimeMixSeq_56590489092147
// MI455X (gfx1250) — compile-verified
//
#include <hip/hip_runtime.h>
#include <cstdint>

// Problem constants (match reference)
#define B_ 16
#define T_ 2048
#define D_ 1024

// Tiling
#define CH 8                 // time-chunks per block == waves per block
#define L_ (T_ / CH)         // 256 time steps per chunk
#define TS 8                 // time steps per LDS tile
#define NTILE (L_ / TS)      // 32 tiles per chunk
#define DT 32                // channels per block (one lane per channel)
#define NTHREADS (CH * DT)   // 256

typedef unsigned int u32;
typedef unsigned long long u64;

#ifndef __has_builtin
#define __has_builtin(x) 0
#endif

#if __has_builtin(__builtin_amdgcn_global_load_async_to_lds_b32)
#define HAVE_ASYNC_BUILTIN 1
#else
#define HAVE_ASYNC_BUILTIN 0
#endif

// Builtin prototype takes address-space-qualified int pointers:
//   (AS1 int* global_src, AS3 int* lds_dst, imm offset, imm cpol)
typedef __attribute__((address_space(1))) int gint_t;
typedef __attribute__((address_space(3))) int lint_t;

// One async 4B/lane copy: per-lane global addr -> per-lane LDS addr.
// A full wave issues 32 consecutive d-channels => one coalesced 128B line.
__device__ __forceinline__ void async_ld_b32(const float* gp, float* lp) {
#if HAVE_ASYNC_BUILTIN
  __builtin_amdgcn_global_load_async_to_lds_b32(
      (gint_t*)(uintptr_t)gp,
      (lint_t*)(u32)(uintptr_t)lp,
      0, 0);
#else
  u32 lo = (u32)(uintptr_t)lp;
  u64 ga = (u64)(uintptr_t)gp;
  asm volatile("global_load_async_to_lds_b32 %0, %1, off"
               :: "v"(lo), "v"(ga) : "memory");
#endif
}

// s_wait_asynccnt with a compiler memory barrier so LDS reads can't be
// hoisted above the wait.
template <int N>
__device__ __forceinline__ void wait_asynccnt() {
  asm volatile("s_wait_asynccnt %0" :: "i"(N) : "memory");
}

// Stream one chunk (L_ steps) of k/v/w through double-buffered async LDS
// tiles. Computes the chunk-affine accumulators:
//   W  = prod exp(w_t)
//   Ua = sum (decayed) exp(k_t)
//   Ub = sum (decayed) exp(k_t)*v_t
// With WRITE: also emits out_t = (W_t*b0 + Ub_t) / (W_t*a0 + Ua_t).
template <bool WRITE>
__device__ __forceinline__ void stream_chunk(
    const float* __restrict__ pk, const float* __restrict__ pv,
    const float* __restrict__ pw, float* __restrict__ pout,
    float a0, float b0, float* sT, int tid,
    float& Wo, float& Uao, float& Ubo) {
  // LDS slot: [buf q][array arr][ts][tid]
  auto slot = [&](int q, int arr, int ts) -> float* {
    return sT + (((q * 3 + arr) * TS + ts) * NTHREADS + tid);
  };
  auto issue = [&](int g, int q) {
#pragma unroll
    for (int ts = 0; ts < TS; ++ts) {
      size_t off = (size_t)(g * TS + ts) * D_;
      async_ld_b32(pk + off, slot(q, 0, ts));
      async_ld_b32(pv + off, slot(q, 1, ts));
      async_ld_b32(pw + off, slot(q, 2, ts));
    }
  };

  float W = 1.f, Ua = 0.f, Ub = 0.f;
  issue(0, 0);
  for (int g = 0; g < NTILE; ++g) {
    const int q = g & 1;
    if (g + 1 < NTILE) {
      issue(g + 1, q ^ 1);          // prefetch next tile into other buffer
      wait_asynccnt<3 * TS>();      // wait until current tile (older 24) landed
    } else {
      wait_asynccnt<0>();
    }
#pragma unroll
    for (int ts = 0; ts < TS; ++ts) {
      float kk = *slot(q, 0, ts);
      float vv = *slot(q, 1, ts);
      float ww = *slot(q, 2, ts);
      float ew = __expf(ww);
      float ek = __expf(kk);
      W *= ew;
      Ua = fmaf(ew, Ua, ek);
      Ub = fmaf(ew, Ub, ek * vv);
      if (WRITE) {
        float av = fmaf(W, a0, Ua);
        float bv = fmaf(W, b0, Ub);
        __builtin_nontemporal_store(__fdividef(bv, av),
                                    pout + (size_t)(g * TS + ts) * D_);
      }
    }
  }
  Wo = W; Uao = Ua; Ubo = Ub;
}

__global__ __launch_bounds__(NTHREADS)
void TimeMixSeq_56590489092147_kernel(const float* __restrict__ k,
                                      const float* __restrict__ v,
                                      const float* __restrict__ w,
                                      float* __restrict__ out) {
  __shared__ float sTile[2][3][TS][NTHREADS];          // 48 KB stream buffers
  __shared__ float sW[CH][DT], sUa[CH][DT], sUb[CH][DT];
  __shared__ float sA0[CH][DT], sB0[CH][DT];

  const int tid   = threadIdx.x;
  const int lane  = tid & (DT - 1);   // channel within block
  const int chunk = tid >> 5;         // time chunk == wave id
  const int bb    = blockIdx.x >> 5;  // D_/DT = 32 blocks per batch
  const int d0    = (blockIdx.x & 31) * DT;

  const size_t base =
      ((size_t)bb * T_ + (size_t)chunk * L_) * D_ + (size_t)(d0 + lane);
  const float* pk = k + base;
  const float* pv = v + base;
  const float* pw = w + base;
  float* pout = out + base;
  float* sT = &sTile[0][0][0][0];

  // ---- Pass 1: per-chunk affine accumulators (parallel over T) ----
  float W, Ua, Ub;
  stream_chunk<false>(pk, pv, pw, nullptr, 0.f, 0.f, sT, tid, W, Ua, Ub);
  sW[chunk][lane] = W;
  sUa[chunk][lane] = Ua;
  sUb[chunk][lane] = Ub;
  __syncthreads();

  // ---- Pass 2: tiny serial scan over 8 chunks (wave 0 only) ----
  if (tid < DT) {
    float A = 0.f, Bv = 0.f;
#pragma unroll
    for (int c = 0; c < CH; ++c) {
      sA0[c][tid] = A;
      sB0[c][tid] = Bv;
      A  = fmaf(sW[c][tid], A,  sUa[c][tid]);
      Bv = fmaf(sW[c][tid], Bv, sUb[c][tid]);
    }
  }
  __syncthreads();

  // ---- Pass 3: replay each chunk from its start state, write out ----
  const float a0 = sA0[chunk][lane];
  const float b0 = sB0[chunk][lane];
  stream_chunk<true>(pk, pv, pw, pout, a0, b0, sT, tid, W, Ua, Ub);
}

extern "C" void kernel_launch(void* const* d_in, const int* in_sizes, int n_in,
                              void* d_out, int out_size, void* d_ws, size_t ws_size,
                              hipStream_t stream) {
  (void)in_sizes; (void)n_in; (void)out_size; (void)d_ws; (void)ws_size;
  const float* k = (const float*)d_in[0];
  const float* v = (const float*)d_in[1];
  const float* w = (const float*)d_in[2];
  float* out = (float*)d_out;

  dim3 grid(B_ * (D_ / DT));   // 512 blocks
  dim3 block(NTHREADS);        // 256 threads = 8 wave32
  TimeMixSeq_56590489092147_kernel<<<grid, block, 0, stream>>>(k, v, w, out);
}